// P4ConvEncoder_13984413515807
// MI455X (gfx1250) — compile-verified
//
#include <hip/hip_runtime.h>
#include <hip/hip_bf16.h>

// ---------------------------------------------------------------------------
// P4Conv encoder for MI455X (gfx1250, wave32).
//
// Algebra:  max_k [ g·W_f + (g-a)·W_d3 + d*W_d4 ]
//         = max_k [ g·(W_f+W_d3) ] - a·W_d3 + d*W_d4
// PF[frame,n,o] = xyz[n]·Wc[o] (Wc = W_f + W_d[:, :3]) is computed once per
// frame with V_WMMA_F32_16X16X4_F32 (K=3 padded to 4, exact fp32); the
// grouped k-max becomes a gather-max over PF rows selected by the ball query
// (first NSAMPLES in-radius indices ascending, via wave32 ballot).
// Encode kernel stages neighbor xyz into LDS via GLOBAL_LOAD_ASYNC_TO_LDS_B128
// (ASYNCcnt-tracked).
// ---------------------------------------------------------------------------

#define B_   4
#define T_   8
#define N_   1024
#define M_   512        // N / SPATIAL_STRIDE
#define DIM_ 128
#define NS_  32         // NSAMPLES
#define BT_  (B_ * T_)
#define R2_  0.49f      // RADIUS^2
#define NEG_INF_ (-3.402823466e+38f)

typedef __attribute__((ext_vector_type(2))) float v2f;
typedef __attribute__((ext_vector_type(8))) float v8f;
typedef int v4i_ __attribute__((vector_size(16)));   // matches builtin param type

#define HAS_ASYNC_LDS 0
#if defined(__has_builtin)
#if __has_builtin(__builtin_amdgcn_global_load_async_to_lds_b128) && \
    __has_builtin(__builtin_amdgcn_s_wait_asynccnt)
#undef HAS_ASYNC_LDS
#define HAS_ASYNC_LDS 1
#endif
#endif

// ---------------------------------------------------------------------------
// Kernel 1: farthest point sampling, one block per (b,t) frame.
// 256 threads, 4 points/thread in registers. Matches JAX semantics:
// mind=1e10, last=0, 511 steps of {mind=min(mind,d2(last)); last=argmax(mind)}
// with first-index tie-break. Also gathers anchor xyz.
// ---------------------------------------------------------------------------
__global__ __launch_bounds__(256) void fps_kernel(
    const float* __restrict__ x,        // (B,T,N,4)
    int*         __restrict__ a_idx,    // (BT,M)
    float*       __restrict__ anchor)   // (BT,M,3)
{
    const int f   = blockIdx.x;
    const int tid = threadIdx.x;
    const float* xb = x + (size_t)f * N_ * 4;

    __shared__ float s_val[256];
    __shared__ int   s_idx[256];
    __shared__ float s_pt[3];
    __shared__ int   s_last;

    float px[4], py[4], pz[4], md[4];
    int   pn[4];
#pragma unroll
    for (int j = 0; j < 4; ++j) {
        int n = tid + 256 * j;
        pn[j] = n;
        px[j] = xb[n * 4 + 0];
        py[j] = xb[n * 4 + 1];
        pz[j] = xb[n * 4 + 2];
        md[j] = 1e10f;
    }
    if (tid == 0) { a_idx[f * M_] = 0; s_last = 0; }
    __syncthreads();

    for (int m = 1; m < M_; ++m) {
        int last = s_last;
        if ((last & 255) == tid) {            // owner broadcasts coords
            int j = last >> 8;
            s_pt[0] = px[j]; s_pt[1] = py[j]; s_pt[2] = pz[j];
        }
        __syncthreads();
        float lx = s_pt[0], ly = s_pt[1], lz = s_pt[2];

        float bv = -1.0f; int bi = N_;
#pragma unroll
        for (int j = 0; j < 4; ++j) {
            float dx = px[j] - lx, dy = py[j] - ly, dz = pz[j] - lz;
            float d2 = dx * dx + dy * dy + dz * dz;
            md[j] = fminf(md[j], d2);
            if (md[j] > bv || (md[j] == bv && pn[j] < bi)) { bv = md[j]; bi = pn[j]; }
        }
        s_val[tid] = bv; s_idx[tid] = bi;
        __syncthreads();
        for (int s = 128; s > 0; s >>= 1) {
            if (tid < s) {
                float v2 = s_val[tid + s]; int i2 = s_idx[tid + s];
                if (v2 > s_val[tid] || (v2 == s_val[tid] && i2 < s_idx[tid])) {
                    s_val[tid] = v2; s_idx[tid] = i2;
                }
            }
            __syncthreads();
        }
        if (tid == 0) { s_last = s_idx[0]; a_idx[f * M_ + m] = s_idx[0]; }
        __syncthreads();
    }

    for (int m = tid; m < M_; m += 256) {
        int idx = a_idx[f * M_ + m];
        anchor[(size_t)(f * M_ + m) * 3 + 0] = xb[idx * 4 + 0];
        anchor[(size_t)(f * M_ + m) * 3 + 1] = xb[idx * 4 + 1];
        anchor[(size_t)(f * M_ + m) * 3 + 2] = xb[idx * 4 + 2];
    }
}

// ---------------------------------------------------------------------------
// Kernel 2: PF[f,n,o] = xyz[f,n,:]·(W_f+W_d3)[o,:] via v_wmma_f32_16x16x4_f32.
// One wave per 16-row stripe; the A operand (16 points) stays in registers
// across all 8 column tiles -> 8 back-to-back WMMAs per wave.
// Operand construction is branchless (v_cndmask on lane-half) so EXEC stays
// all-ones around the WMMAs, per the ISA restriction.
// Lane layouts per ISA 7.12.2:
//   A(16x4,f32): lane<16 -> {K0,K1} of row M=lane; lane>=16 -> {K2,K3}
//   B(4x16)    : mirrored (lane<16 -> rows K0/K1, col=lane; lane>=16 -> K2/K3)
//   C/D(16x16) : vgpr j -> row j (lanes 0-15) / row j+8 (lanes 16-31), col=lane&15
// ---------------------------------------------------------------------------
__global__ __launch_bounds__(32) void point_feat_wmma(
    const float* __restrict__ x,     // (B,T,N,4)
    const float* __restrict__ W_f,   // (DIM,3)
    const float* __restrict__ W_d,   // (DIM,4)
    float*       __restrict__ PF)    // (BT,N,DIM)
{
    const int f    = blockIdx.x;           // frame
    const int rt   = blockIdx.y;           // row tile (N/16)
    const int lane = threadIdx.x;
    const int ml   = lane & 15;
    const bool hi  = (lane & 16) != 0;     // lane half selects K pair

    // A operand: point row (x,y,z,0) -- loaded once, reused for 8 WMMAs
    const int row = rt * 16 + ml;
    const float4 p = *(const float4*)(x + ((size_t)f * N_ + row) * 4);
    v2f a;
    a.x = hi ? p.z : p.x;
    a.y = hi ? 0.0f : p.y;

    float* pfb = PF + ((size_t)f * N_ + (size_t)rt * 16) * DIM_;

#pragma unroll
    for (int ct = 0; ct < DIM_ / 16; ++ct) {
        // B operand: Wc^T column o = ct*16+ml, Wc = W_f + W_d[:, :3], K padded
        const int o = ct * 16 + ml;
        const float c0 = W_f[o * 3 + 0] + W_d[o * 4 + 0];
        const float c1 = W_f[o * 3 + 1] + W_d[o * 4 + 1];
        const float c2 = W_f[o * 3 + 2] + W_d[o * 4 + 2];
        v2f b;
        b.x = hi ? c2 : c0;
        b.y = hi ? 0.0f : c1;

        v8f acc = {};
        acc = __builtin_amdgcn_wmma_f32_16x16x4_f32(
            /*neg_a=*/false, a, /*neg_b=*/false, b,
            /*c_mod=*/(short)0, acc, /*reuse_a=*/false, /*reuse_b=*/false);

        const int rbase = hi ? 8 : 0;
#pragma unroll
        for (int j = 0; j < 8; ++j)
            pfb[(size_t)(j + rbase) * DIM_ + ct * 16 + ml] = acc[j];
    }
}

// ---------------------------------------------------------------------------
// Kernel 3: ball query + gather-max + temporal max + positional embed + relu.
// One block per (b,t); 3 clamped neighbor frames' xyz in LDS (48 KB, float4
// records filled with async-to-LDS b128 copies).
// Wave w handles anchors m = w, w+8, ... Each lane owns 4 output channels.
// Selection = first NS_ in-radius indices ascending (wave32 ballot + ctz),
// exactly the reference's sort-by-index-take-first-32 semantics.
// ---------------------------------------------------------------------------
__global__ __launch_bounds__(256) void encode_kernel(
    const float* __restrict__ x,       // (B,T,N,4)
    const float* __restrict__ W_d,     // (DIM,4)
    const float* __restrict__ W_p,     // (DIM,4)
    const float* __restrict__ b_p,     // (DIM)
    const float* __restrict__ anchor,  // (BT,M,3)
    const float* __restrict__ PF,      // (BT,N,DIM)
    float*       __restrict__ out)     // (B,T*M,DIM)
{
    const int f = blockIdx.x;
    const int b = f / T_;
    const int t = f % T_;

    __shared__ float4 spts[3][N_];     // (x,y,z,w) per point, 16B records

    for (int d = 0; d < 3; ++d) {
        int tc = t + d - 1;
        tc = tc < 0 ? 0 : (tc > T_ - 1 ? T_ - 1 : tc);
        const float4* xb = (const float4*)(x + (size_t)(b * T_ + tc) * N_ * 4);
        for (int n = threadIdx.x; n < N_; n += 256) {
#if HAS_ASYNC_LDS
            __builtin_amdgcn_global_load_async_to_lds_b128(
                (v4i_ __attribute__((address_space(1)))*)(xb + n),
                (v4i_ __attribute__((address_space(3)))*)&spts[d][n],
                /*offset=*/0, /*cpol=*/0);
#else
            spts[d][n] = xb[n];
#endif
        }
    }
#if HAS_ASYNC_LDS
    __builtin_amdgcn_s_wait_asynccnt(0);
#endif
    __syncthreads();

    const int w    = threadIdx.x >> 5;
    const int lane = threadIdx.x & 31;
    const int co   = lane * 4;              // this lane's 4 output channels

    // per-lane weights (invariant over anchors)
    float wcp0[4], wcp1[4], wcp2[4], wd4[4], wp4[4], bpv[4];
#pragma unroll
    for (int j = 0; j < 4; ++j) {
        int o = co + j;
        float d0 = W_d[o * 4 + 0], d1 = W_d[o * 4 + 1], d2w = W_d[o * 4 + 2];
        wd4[j]  = W_d[o * 4 + 3];
        wcp0[j] = W_p[o * 4 + 0] - d0;      // a·(Wp3 - Wd3) folds pos + feat const
        wcp1[j] = W_p[o * 4 + 1] - d1;
        wcp2[j] = W_p[o * 4 + 2] - d2w;
        wp4[j]  = W_p[o * 4 + 3];
        bpv[j]  = b_p[o];
    }

    const float tcoord = (float)(t + 1);

    for (int m = w; m < M_; m += 8) {
        const float* av = anchor + (size_t)(f * M_ + m) * 3;
        const float ax = av[0], ay = av[1], az = av[2];

        float best[4] = { NEG_INF_, NEG_INF_, NEG_INF_, NEG_INF_ };

        for (int d = 0; d < 3; ++d) {
            int tc = t + d - 1;
            tc = tc < 0 ? 0 : (tc > T_ - 1 ? T_ - 1 : tc);
            const float* pfb = PF + (size_t)(b * T_ + tc) * N_ * DIM_;

            float md4[4] = { NEG_INF_, NEG_INF_, NEG_INF_, NEG_INF_ };
            int count = 0;
            bool any = false;

            for (int base = 0; base < N_ && count < NS_; base += 32) {
                int n = base + lane;
                float4 q = spts[d][n];
                float dx = q.x - ax;
                float dy = q.y - ay;
                float dz = q.z - az;
                float d2 = dx * dx + dy * dy + dz * dz;
                unsigned msk = __builtin_amdgcn_ballot_w32(d2 < R2_);
                while (msk && count < NS_) {
                    int bit = __builtin_ctz(msk);
                    msk &= msk - 1;
                    const float4 r =
                        *(const float4*)(pfb + (size_t)(base + bit) * DIM_ + co);
                    md4[0] = fmaxf(md4[0], r.x);
                    md4[1] = fmaxf(md4[1], r.y);
                    md4[2] = fmaxf(md4[2], r.z);
                    md4[3] = fmaxf(md4[3], r.w);
                    ++count; any = true;
                }
            }
            if (!any) {   // reference falls back to neighbor index 0
                const float4 r = *(const float4*)(pfb + co);
                md4[0] = r.x; md4[1] = r.y; md4[2] = r.z; md4[3] = r.w;
            }
            const float doff = (float)(d - 1);
#pragma unroll
            for (int j = 0; j < 4; ++j)
                best[j] = fmaxf(best[j], md4[j] + doff * wd4[j]);
        }

        float* op = out + ((size_t)f * M_ + m) * DIM_ + co;
        float4 res;
        float v0 = best[0] + ax * wcp0[0] + ay * wcp1[0] + az * wcp2[0] + tcoord * wp4[0] + bpv[0];
        float v1 = best[1] + ax * wcp0[1] + ay * wcp1[1] + az * wcp2[1] + tcoord * wp4[1] + bpv[1];
        float v2 = best[2] + ax * wcp0[2] + ay * wcp1[2] + az * wcp2[2] + tcoord * wp4[2] + bpv[2];
        float v3 = best[3] + ax * wcp0[3] + ay * wcp1[3] + az * wcp2[3] + tcoord * wp4[3] + bpv[3];
        res.x = v0 > 0.f ? v0 : 0.f;
        res.y = v1 > 0.f ? v1 : 0.f;
        res.z = v2 > 0.f ? v2 : 0.f;
        res.w = v3 > 0.f ? v3 : 0.f;
        *(float4*)op = res;
    }
}

// ---------------------------------------------------------------------------
extern "C" void kernel_launch(void* const* d_in, const int* in_sizes, int n_in,
                              void* d_out, int out_size, void* d_ws, size_t ws_size,
                              hipStream_t stream) {
    const float* x   = (const float*)d_in[0];
    const float* W_d = (const float*)d_in[1];
    const float* W_f = (const float*)d_in[2];
    const float* W_p = (const float*)d_in[3];
    const float* b_p = (const float*)d_in[4];
    float* out = (float*)d_out;

    // workspace layout: [a_idx: BT*M ints][anchor: BT*M*3 f32][PF: BT*N*DIM f32]
    int*   a_idx  = (int*)d_ws;
    float* anchor = (float*)((char*)d_ws + (size_t)BT_ * M_ * sizeof(int));
    float* PF     = anchor + (size_t)BT_ * M_ * 3;

    fps_kernel<<<BT_, 256, 0, stream>>>(x, a_idx, anchor);

    dim3 g2(BT_, N_ / 16);
    point_feat_wmma<<<g2, 32, 0, stream>>>(x, W_f, W_d, PF);

    encode_kernel<<<BT_, 256, 0, stream>>>(x, W_d, W_p, b_p, anchor, PF, out);
}